// GatedAttentionPool_36455682408887
// MI455X (gfx1250) — compile-verified
//
#include <hip/hip_runtime.h>
#include <hip/hip_bf16.h>
#include <math.h>

typedef __attribute__((ext_vector_type(2))) float v2f;
typedef __attribute__((ext_vector_type(8))) float v8f;

#define SDIM 1024   // 32*32 spatial
#define CDIM 256
#define RCD  32
#define KVN  768    // int(1024 * 0.75)

__device__ __forceinline__ float dsigmoid(float x) { return 1.0f / (1.0f + expf(-x)); }

// ---------------------------------------------------------------------------
// K1: 2x2 max/avg pool + mix  ->  combined[4][256][1024]
// ---------------------------------------------------------------------------
__global__ void pool_mix_kernel(const float* __restrict__ x,
                                const float* __restrict__ mix_param,
                                float* __restrict__ combined) {
    int idx = blockIdx.x * blockDim.x + threadIdx.x;     // over 2*256*1024
    if (idx >= 2 * CDIM * SDIM) return;
    int s = idx & (SDIM - 1);
    int c = (idx >> 10) & (CDIM - 1);
    int b = idx >> 18;
    int w = s & 31, h = s >> 5;
    const float* xp = x + (((size_t)(b * CDIM + c)) * 64 + h * 2) * 64 + w * 2;
    float a0 = xp[0], a1 = xp[1], a2 = xp[64], a3 = xp[65];
    float mx = fmaxf(fmaxf(a0, a1), fmaxf(a2, a3));
    float av = (a0 + a1 + a2 + a3) * 0.25f;
    float mw = dsigmoid(mix_param[0]);
    float x1 = mx + mw * (av - mx);
    float x2 = av + mw * (mx - av);
    combined[((size_t)(b * 2 + 0) * CDIM + c) * SDIM + s] = x1;
    combined[((size_t)(b * 2 + 1) * CDIM + c) * SDIM + s] = x2;
}

// ---------------------------------------------------------------------------
// K2a: spatial mean per (b2,c) -> pooled[4*256]
// ---------------------------------------------------------------------------
__global__ void pool_mean_kernel(const float* __restrict__ combined,
                                 float* __restrict__ pooled) {
    __shared__ float red[256];
    int bc = blockIdx.x;  // b2*256 + c
    const float* p = combined + (size_t)bc * SDIM;
    int t = threadIdx.x;
    float s = p[t] + p[t + 256] + p[t + 512] + p[t + 768];
    red[t] = s;
    __syncthreads();
    for (int off = 128; off > 0; off >>= 1) {
        if (t < off) red[t] += red[t + off];
        __syncthreads();
    }
    if (t == 0) pooled[bc] = red[0] * (1.0f / (float)SDIM);
}

// ---------------------------------------------------------------------------
// K2b: gating MLP (4x256 -> relu 128 -> sigmoid 1 -> 0/1 gate[4])
// ---------------------------------------------------------------------------
__global__ void gate_mlp_kernel(const float* __restrict__ pooled,
                                const float* __restrict__ w_s1,
                                const float* __restrict__ b_s1,
                                const float* __restrict__ w_s2,
                                const float* __restrict__ b_s2,
                                float* __restrict__ gate) {
    __shared__ float h1[4][128];
    int j = threadIdx.x;  // 0..127
    for (int b2 = 0; b2 < 4; ++b2) {
        float acc = b_s1[j];
        const float* pp = pooled + b2 * CDIM;
        const float* wp = w_s1 + j * CDIM;
        for (int c = 0; c < CDIM; ++c) acc += pp[c] * wp[c];
        h1[b2][j] = fmaxf(acc, 0.0f) * w_s2[j];  // fold second-layer weight in
    }
    __syncthreads();
    if (j < 4) {
        float acc = b_s2[0];
        for (int t = 0; t < 128; ++t) acc += h1[j][t];
        gate[j] = (dsigmoid(acc) > 0.5f) ? 1.0f : 0.0f;
    }
}

// ---------------------------------------------------------------------------
// K3: out[b,c,s] = max(g0*combined[2b], g1*combined[2b+1])
// ---------------------------------------------------------------------------
__global__ void gated_max_kernel(const float* __restrict__ combined,
                                 const float* __restrict__ gate,
                                 float* __restrict__ out) {
    int idx = blockIdx.x * blockDim.x + threadIdx.x;  // over 2*256*1024
    if (idx >= 2 * CDIM * SDIM) return;
    int s = idx & (SDIM - 1);
    int c = (idx >> 10) & (CDIM - 1);
    int b = idx >> 18;
    float g0 = gate[2 * b], g1 = gate[2 * b + 1];
    float v0 = g0 * combined[((size_t)(2 * b) * CDIM + c) * SDIM + s];
    float v1 = g1 * combined[((size_t)(2 * b + 1) * CDIM + c) * SDIM + s];
    out[idx] = fmaxf(v0, v1);
}

// ---------------------------------------------------------------------------
// K4a: channel reduce 256 -> 32 : xf[b2][s][r]
// ---------------------------------------------------------------------------
__global__ void reduce_c_kernel(const float* __restrict__ combined,
                                const float* __restrict__ w_red,
                                const float* __restrict__ b_red,
                                float* __restrict__ xf) {
    int idx = blockIdx.x * blockDim.x + threadIdx.x;  // over 4*1024*32
    if (idx >= 4 * SDIM * RCD) return;
    int r = idx & (RCD - 1);
    int s = (idx >> 5) & (SDIM - 1);
    int b2 = idx >> 15;
    float acc = b_red[r];
    const float* cb = combined + (size_t)b2 * CDIM * SDIM + s;
    const float* wr = w_red + r * CDIM;
    for (int c = 0; c < CDIM; ++c) acc += cb[(size_t)c * SDIM] * wr[c];
    xf[idx] = acc;
}

// ---------------------------------------------------------------------------
// K4b: qkv projection 32 -> 96 split into q/k/v [b2][s][32]
// ---------------------------------------------------------------------------
__global__ void qkv_kernel(const float* __restrict__ xf,
                           const float* __restrict__ w_qkv,
                           const float* __restrict__ b_qkv,
                           float* __restrict__ q, float* __restrict__ k,
                           float* __restrict__ v) {
    int idx = blockIdx.x * blockDim.x + threadIdx.x;  // over 4*1024*96
    if (idx >= 4 * SDIM * 96) return;
    int i = idx % 96;
    int bs = idx / 96;  // b2*1024 + s
    float acc = b_qkv[i];
    const float* xp = xf + (size_t)bs * RCD;
    const float* wp = w_qkv + i * RCD;
    for (int j = 0; j < RCD; ++j) acc += xp[j] * wp[j];
    float* dst = (i < 32) ? q : (i < 64) ? k : v;
    dst[(size_t)bs * RCD + (i & 31)] = acc;
}

// ---------------------------------------------------------------------------
// K5: attention core. One wave per 16-row stripe of the score matrix.
//   scores (WMMA fp32 16x16x4) -> LDS -> exact radix top-k threshold ->
//   masked softmax -> P @ V (WMMA fp32 16x16x4)
// grid = (4 batches, 64 row tiles), block = 32 (one wave), dynLDS = 64 KB
// ---------------------------------------------------------------------------
__global__ void __launch_bounds__(32)
attn_core_kernel(const float* __restrict__ qarr, const float* __restrict__ karr,
                 const float* __restrict__ varr, float* __restrict__ oarr) {
    extern __shared__ float smem[];  // 16 * 1024 floats = 64 KB
    const int b2 = blockIdx.x;
    const int mt = blockIdx.y;
    const int lane = threadIdx.x;
    const int hi = lane >> 4;   // 0 or 1 (lane half)
    const int l16 = lane & 15;

    const float* Q = qarr + (size_t)b2 * SDIM * RCD;
    const float* K = karr + (size_t)b2 * SDIM * RCD;
    const float* V = varr + (size_t)b2 * SDIM * RCD;
    float* O = oarr + (size_t)b2 * SDIM * RCD;

    // ---- A fragments of the Q tile (16 rows x rc=32, 8 chunks of K=4) ----
    v2f afrag[8];
    {
        const int m = mt * 16 + l16;
        for (int kk = 0; kk < 8; ++kk) {
            int r = kk * 4 + hi * 2;
            afrag[kk].x = Q[(size_t)m * RCD + r];
            afrag[kk].y = Q[(size_t)m * RCD + r + 1];
        }
    }
    const float scale = 0.1767766952966369f;  // 1/sqrt(32)

    // ---- scores: S[16][1024] into LDS via fp32 WMMA ----
    for (int tt = 0; tt < SDIM / 16; ++tt) {
        v8f c = {};
        const int t = tt * 16 + l16;
        for (int kk = 0; kk < 8; ++kk) {
            int r = kk * 4 + hi * 2;
            v2f bfrag;
            bfrag.x = K[(size_t)t * RCD + r];      // B[r][t] = K[t][r]
            bfrag.y = K[(size_t)t * RCD + r + 1];
            c = __builtin_amdgcn_wmma_f32_16x16x4_f32(
                false, afrag[kk], false, bfrag, (short)0, c, false, false);
        }
        for (int vv = 0; vv < 8; ++vv) {
            float s = c[vv] * scale;
            if (__builtin_isnan(s)) s = -__builtin_inff();
            smem[(vv + 8 * hi) * SDIM + tt * 16 + l16] = s;
        }
    }
    __syncthreads();

    // ---- per-row exact top-KVN threshold (bit-descent radix select),
    //      then masked softmax written back into smem as P ----
    for (int row = 0; row < 16; ++row) {
        float vals[32];
        unsigned keys[32];
        float vmax = -__builtin_inff();
        for (int j = 0; j < 32; ++j) {
            float f = smem[row * SDIM + j * 32 + lane];
            vals[j] = f;
            unsigned u = __float_as_uint(f);
            keys[j] = (u & 0x80000000u) ? ~u : (u | 0x80000000u);  // order-preserving
            vmax = fmaxf(vmax, f);
        }
        for (int off = 16; off > 0; off >>= 1)
            vmax = fmaxf(vmax, __shfl_xor(vmax, off, 32));

        unsigned X = 0u;  // largest X with count(key >= X) >= KVN == KVN-th largest key
        for (int bit = 31; bit >= 0; --bit) {
            unsigned cand = X | (1u << bit);
            int cnt = 0;
            for (int j = 0; j < 32; ++j) cnt += (keys[j] >= cand) ? 1 : 0;
            for (int off = 16; off > 0; off >>= 1)
                cnt += __shfl_xor(cnt, off, 32);
            if (cnt >= KVN) X = cand;
        }

        float se = 0.0f;
        for (int j = 0; j < 32; ++j)
            if (keys[j] >= X) se += expf(vals[j] - vmax);
        for (int off = 16; off > 0; off >>= 1)
            se += __shfl_xor(se, off, 32);
        float inv = 1.0f / se;
        for (int j = 0; j < 32; ++j) {
            float p = (keys[j] >= X) ? expf(vals[j] - vmax) * inv : 0.0f;
            smem[row * SDIM + j * 32 + lane] = p;
        }
    }
    __syncthreads();

    // ---- out tile (16 x 32) = P(16x1024) @ V(1024x32), fp32 WMMA ----
    v8f acc0 = {};
    v8f acc1 = {};
    for (int tt = 0; tt < SDIM / 4; ++tt) {
        const int tcol = tt * 4 + hi * 2;
        v2f a;
        a.x = smem[l16 * SDIM + tcol];
        a.y = smem[l16 * SDIM + tcol + 1];
        v2f b0, b1;
        b0.x = V[(size_t)tcol * RCD + l16];
        b0.y = V[(size_t)(tcol + 1) * RCD + l16];
        b1.x = V[(size_t)tcol * RCD + 16 + l16];
        b1.y = V[(size_t)(tcol + 1) * RCD + 16 + l16];
        acc0 = __builtin_amdgcn_wmma_f32_16x16x4_f32(
            false, a, false, b0, (short)0, acc0, false, false);
        acc1 = __builtin_amdgcn_wmma_f32_16x16x4_f32(
            false, a, false, b1, (short)0, acc1, false, false);
    }
    for (int vv = 0; vv < 8; ++vv) {
        int row = vv + 8 * hi;
        O[(size_t)(mt * 16 + row) * RCD + l16] = acc0[vv];
        O[(size_t)(mt * 16 + row) * RCD + 16 + l16] = acc1[vv];
    }
}

// ---------------------------------------------------------------------------
// K6: expand channels 32 -> 256 : attn_out[b2][c][s]
// ---------------------------------------------------------------------------
__global__ void expand_c_kernel(const float* __restrict__ oattn,
                                const float* __restrict__ w_red,
                                float* __restrict__ attn_out) {
    int idx = blockIdx.x * blockDim.x + threadIdx.x;  // over 4*256*1024
    if (idx >= 4 * CDIM * SDIM) return;
    int s = idx & (SDIM - 1);
    int c = (idx >> 10) & (CDIM - 1);
    int b2 = idx >> 18;
    const float* op = oattn + ((size_t)b2 * SDIM + s) * RCD;
    float acc = 0.0f;
    for (int r = 0; r < RCD; ++r) acc += op[r] * w_red[r * CDIM + c];
    attn_out[idx] = acc;
}

// ---------------------------------------------------------------------------
extern "C" void kernel_launch(void* const* d_in, const int* in_sizes, int n_in,
                              void* d_out, int out_size, void* d_ws, size_t ws_size,
                              hipStream_t stream) {
    const float* x         = (const float*)d_in[0];
    const float* mix_param = (const float*)d_in[1];
    const float* w_red     = (const float*)d_in[2];
    const float* b_red     = (const float*)d_in[3];
    const float* w_qkv     = (const float*)d_in[4];
    const float* b_qkv     = (const float*)d_in[5];
    const float* w_s1      = (const float*)d_in[6];
    const float* b_s1      = (const float*)d_in[7];
    const float* w_s2      = (const float*)d_in[8];
    const float* b_s2      = (const float*)d_in[9];
    float* out = (float*)d_out;                       // [2*256*1024] then [4*256*1024]
    float* ws  = (float*)d_ws;

    // workspace layout (floats)
    float* combined = ws;                              // 4*256*1024
    float* pooled   = combined + 4 * CDIM * SDIM;      // 1024
    float* gate     = pooled + 4 * CDIM;               // 4
    float* xf       = gate + 4;                        // 4*1024*32
    float* qb       = xf + 4 * SDIM * RCD;
    float* kb       = qb + 4 * SDIM * RCD;
    float* vb       = kb + 4 * SDIM * RCD;
    float* oattn    = vb + 4 * SDIM * RCD;

    // pool + mix
    pool_mix_kernel<<<(2 * CDIM * SDIM) / 256, 256, 0, stream>>>(x, mix_param, combined);

    // gating branch
    pool_mean_kernel<<<4 * CDIM, 256, 0, stream>>>(combined, pooled);
    gate_mlp_kernel<<<1, 128, 0, stream>>>(pooled, w_s1, b_s1, w_s2, b_s2, gate);
    gated_max_kernel<<<(2 * CDIM * SDIM) / 256, 256, 0, stream>>>(combined, gate, out);

    // attention branch
    reduce_c_kernel<<<(4 * SDIM * RCD) / 256, 256, 0, stream>>>(combined, w_red, b_red, xf);
    qkv_kernel<<<(4 * SDIM * 96) / 256, 256, 0, stream>>>(xf, w_qkv, b_qkv, qb, kb, vb);
    attn_core_kernel<<<dim3(4, SDIM / 16), 32, 16 * SDIM * sizeof(float), stream>>>(
        qb, kb, vb, oattn);
    expand_c_kernel<<<(4 * CDIM * SDIM) / 256, 256, 0, stream>>>(
        oattn, w_red, out + 2 * CDIM * SDIM);
}